// lossEvaluator_86586540687580
// MI455X (gfx1250) — compile-verified
//
#include <hip/hip_runtime.h>
#include <hip/hip_bf16.h>
#include <math.h>

#define B_DIM 256
#define P_DIM 2048
#define MARGIN_F 0.1f
#define LAMDA2_F 0.1f

typedef __attribute__((ext_vector_type(2))) float v2f;
typedef __attribute__((ext_vector_type(4))) float v4f;
typedef __attribute__((ext_vector_type(8))) float v8f;

// ---------------------------------------------------------------------------
// Kernel 0: init simMax to -inf (for atomic max) and zero the entropy accum.
// ---------------------------------------------------------------------------
__global__ void k_init(float* __restrict__ simMax, float* __restrict__ entroSum) {
    int id = blockIdx.x * blockDim.x + threadIdx.x;
    if (id < B_DIM * B_DIM) simMax[id] = -INFINITY;
    if (id == 0) entroSum[0] = 0.0f;
}

// ---------------------------------------------------------------------------
// Kernel 1: streaming max over P (512 MB pass).
//   grid = (8 p-chunks, 256 i-rows), block = 256 threads.
//   Thread t: column group g = t&63 (4 columns via b128 load),
//             p-subrange sub = t>>6 (64 consecutive p values).
//   Non-temporal b128 loads (read-once stream, don't pollute L2).
//   Also peels off the diagonal slice match[i,p] = s[i,p,i] for free.
// ---------------------------------------------------------------------------
__global__ void k_max(const float* __restrict__ s,
                      float* __restrict__ simMax,
                      float* __restrict__ match) {
    const int i   = blockIdx.y;        // 0..255
    const int pc  = blockIdx.x;        // 0..7
    const int t   = threadIdx.x;       // 0..255
    const int g   = t & 63;            // j = 4g .. 4g+3
    const int sub = t >> 6;            // 0..3
    const int p0  = pc * 256 + sub * 64;

    const v4f* base = reinterpret_cast<const v4f*>(s) + (size_t)i * P_DIM * (B_DIM / 4);
    float* mrow = match + (size_t)i * P_DIM;

    v4f vmax = { -INFINITY, -INFINITY, -INFINITY, -INFINITY };
    const bool diag = (g == (i >> 2));
    const int  dc   = i & 3;

#pragma unroll 4
    for (int k = 0; k < 64; ++k) {
        const int p = p0 + k;
        v4f v = __builtin_nontemporal_load(base + (size_t)p * (B_DIM / 4) + g);
        vmax.x = fmaxf(vmax.x, v.x);
        vmax.y = fmaxf(vmax.y, v.y);
        vmax.z = fmaxf(vmax.z, v.z);
        vmax.w = fmaxf(vmax.w, v.w);
        if (diag) {
            float dv = (dc == 0) ? v.x : (dc == 1) ? v.y : (dc == 2) ? v.z : v.w;
            mrow[p] = dv;
        }
    }

    __shared__ v4f red[256];
    red[t] = vmax;
    __syncthreads();
    if (t < 64) {
        v4f a = red[t], b = red[t + 64], c = red[t + 128], d = red[t + 192];
        float m0 = fmaxf(fmaxf(a.x, b.x), fmaxf(c.x, d.x));
        float m1 = fmaxf(fmaxf(a.y, b.y), fmaxf(c.y, d.y));
        float m2 = fmaxf(fmaxf(a.z, b.z), fmaxf(c.z, d.z));
        float m3 = fmaxf(fmaxf(a.w, b.w), fmaxf(c.w, d.w));
        float* dst = simMax + (size_t)i * B_DIM + t * 4;
        atomicMax(&dst[0], m0);
        atomicMax(&dst[1], m1);
        atomicMax(&dst[2], m2);
        atomicMax(&dst[3], m3);
    }
}

// ---------------------------------------------------------------------------
// Kernel 2: per-row entropy of softmax over match[i, 0:2048].
//   entropy_i = m + log(S) - T/S with S = sum exp(x-m), T = sum exp(x-m)*x.
//   One block per row i; result atomically accumulated.
// ---------------------------------------------------------------------------
__global__ void k_entropy(const float* __restrict__ match,
                          float* __restrict__ entroSum) {
    const int i = blockIdx.x;
    const int t = threadIdx.x;
    const float* m = match + (size_t)i * P_DIM;

    float x[8];
    float lmax = -INFINITY;
#pragma unroll
    for (int k = 0; k < 8; ++k) {
        x[k] = m[t + 256 * k];
        lmax = fmaxf(lmax, x[k]);
    }

    __shared__ float redA[256];
    __shared__ float redB[256];

    redA[t] = lmax;
    __syncthreads();
    for (int off = 128; off > 0; off >>= 1) {
        if (t < off) redA[t] = fmaxf(redA[t], redA[t + off]);
        __syncthreads();
    }
    const float mx = redA[0];
    __syncthreads();

    float S = 0.0f, T = 0.0f;
#pragma unroll
    for (int k = 0; k < 8; ++k) {
        float e = expf(x[k] - mx);
        S += e;
        T += e * x[k];
    }
    redA[t] = S;
    redB[t] = T;
    __syncthreads();
    for (int off = 128; off > 0; off >>= 1) {
        if (t < off) {
            redA[t] += redA[t + off];
            redB[t] += redB[t + off];
        }
        __syncthreads();
    }
    if (t == 0) {
        float Sv = redA[0], Tv = redB[0];
        float ent = mx + logf(Sv) - Tv / Sv;
        atomicAdd(entroSum, ent);
    }
}

// ---------------------------------------------------------------------------
// Kernel 3: hinge losses over simMax[256,256] + final combine.
//   Thread t owns column j = t (coalesced column-major walk over i).
//   Block-level sum reduction finished with V_WMMA_F32_16X16X4_F32:
//   A-vgpr0 = per-lane partials (lanes 0..15 -> A[m,0], lanes 16..31 -> A[m,2]),
//   B = ones  =>  D[m,n] = v[m] + v[m+16]; total = (lane0 + lane16) row sums.
//   WMMA executes under full wave-0 EXEC (t < 32 covers all 32 lanes).
// ---------------------------------------------------------------------------
__global__ void k_final(const float* __restrict__ simMax,
                        const int* __restrict__ lbl,
                        const float* __restrict__ entroSum,
                        float* __restrict__ out) {
    __shared__ float pd[256];
    __shared__ int   sl[256];
    __shared__ float redH[256];
    __shared__ float redM[256];

    const int t = threadIdx.x;
    pd[t] = simMax[(size_t)t * B_DIM + t];  // pos_diag
    sl[t] = lbl[t];
    __syncthreads();

    const float pdj = pd[t];
    const int   lj  = sl[t];
    float hsum = 0.0f, msum = 0.0f;
    for (int i = 0; i < B_DIM; ++i) {
        float v    = simMax[(size_t)i * B_DIM + t];
        float mask = (sl[i] != lj) ? 1.0f : 0.0f;
        float h1 = v - pdj + MARGIN_F;   h1 = h1 > 0.0f ? h1 : 0.0f;  // vs col diag
        float h2 = v - pd[i] + MARGIN_F; h2 = h2 > 0.0f ? h2 : 0.0f;  // vs row diag
        hsum += (h1 + h2) * mask;
        msum += mask;
    }
    redH[t] = hsum;
    redM[t] = msum;
    __syncthreads();

    if (t < 32) {  // wave 0, EXEC all ones
        float h = 0.0f, mm = 0.0f;
#pragma unroll
        for (int k = 0; k < 8; ++k) {
            h  += redH[t + 32 * k];
            mm += redM[t + 32 * k];
        }
        v2f aH    = { h,    0.0f };
        v2f aM    = { mm,   0.0f };
        v2f bOnes = { 1.0f, 1.0f };
        v8f c     = {};
        // D[m,n] = v[m] + v[m+16]  (B all-ones makes the VGPR layout of B moot)
        v8f dH = __builtin_amdgcn_wmma_f32_16x16x4_f32(
            false, aH, false, bOnes, (short)0, c, false, false);
        v8f dM = __builtin_amdgcn_wmma_f32_16x16x4_f32(
            false, aM, false, bOnes, (short)0, c, false, false);
        float ph = dH[0] + dH[1] + dH[2] + dH[3] + dH[4] + dH[5] + dH[6] + dH[7];
        float pm = dM[0] + dM[1] + dM[2] + dM[3] + dM[4] + dM[5] + dM[6] + dM[7];
        redH[t] = ph;   // lane l: sum over its 8 M-rows; total = lane0 + lane16
        redM[t] = pm;
    }
    __syncthreads();

    if (t == 0) {
        float hTot = redH[0] + redH[16];
        float mTot = redM[0] + redM[16];
        float pair_num = 1e-6f + mTot;
        float loss = hTot / pair_num + LAMDA2_F * (entroSum[0] / (float)B_DIM);
        out[0] = loss;
    }
}

// ---------------------------------------------------------------------------
extern "C" void kernel_launch(void* const* d_in, const int* in_sizes, int n_in,
                              void* d_out, int out_size, void* d_ws, size_t ws_size,
                              hipStream_t stream) {
    const float* simMM = (const float*)d_in[0];
    const int*   lbl   = (const int*)d_in[1];
    float*       out   = (float*)d_out;

    // workspace layout: simMax (64K f32) | match (512K f32) | entroSum (1 f32)
    float* simMax   = (float*)d_ws;
    float* match    = simMax + (size_t)B_DIM * B_DIM;
    float* entroSum = match + (size_t)B_DIM * P_DIM;

    k_init   <<<dim3(256), dim3(256), 0, stream>>>(simMax, entroSum);
    k_max    <<<dim3(8, 256), dim3(256), 0, stream>>>(simMM, simMax, match);
    k_entropy<<<dim3(256), dim3(256), 0, stream>>>(match, entroSum);
    k_final  <<<dim3(1), dim3(256), 0, stream>>>(simMax, lbl, entroSum, out);
}